// QuadricGridTorch_34239479283919
// MI455X (gfx1250) — compile-verified
//
#include <hip/hip_runtime.h>
#include <hip/hip_bf16.h>
#include <stdint.h>
#include <math.h>

typedef __attribute__((ext_vector_type(2))) float v2f;
typedef __attribute__((ext_vector_type(8))) float v8f;

// ---------------------------------------------------------------------------
// CDNA5 async global->LDS copy (ASYNCcnt-tracked) + wait
// ---------------------------------------------------------------------------
__device__ __forceinline__ void async_g2l_b32(uint32_t lds_off, const void* gaddr) {
  asm volatile("global_load_async_to_lds_b32 %0, %1, off"
               :: "v"(lds_off), "v"(gaddr)
               : "memory");
}
__device__ __forceinline__ void wait_async0() {
  asm volatile("s_wait_asynccnt 0" ::: "memory");
}

// ---------------------------------------------------------------------------
// Setup kernel: computes the six prefix arrays into ws.
//   ws[0:r)    xA3   ws[r:2r)  yA4   ws[2r:3r)  zA5
//   ws[3r:4r)  xA6   ws[4r:5r) yA6   ws[5r:6r)  zA6
// For reso==128 the cumsums are computed as lower-triangular matmuls on the
// WMMA pipe (V_WMMA_F32_16X16X4_F32): 8 chunks of 16 per sequence, two
// sequences per batch (chunks mapped to the 16 N-columns), K=16 via 4 chained
// K=4 WMMAs, then a scalar carry fix-up across chunk totals.
// One wave (32 threads), EXEC all-ones at every WMMA.
// ---------------------------------------------------------------------------
__global__ void qg_setup_kernel(const float* __restrict__ xL,
                                const float* __restrict__ yL,
                                const float* __restrict__ zL,
                                const float* __restrict__ offv,
                                float* __restrict__ ws, int reso) {
  if (reso != 128) {
    // generic sequential fallback (tiny)
    if (threadIdx.x == 0) {
      const float* L[3] = {xL, yL, zL};
      for (int s = 0; s < 3; ++s) {
        float c = offv[s];
        for (int i = 0; i < reso; ++i) {
          float t = (i == 0) ? 0.0f : 2.0f * (L[s][i - 1] + L[s][i]);
          c += t;
          ws[s * reso + i] = c;
        }
        float c6 = 0.0f;
        for (int i = 0; i < reso; ++i) {
          float sv = (i == 0) ? 0.0f
                              : 3.0f * L[s][i - 1] + L[s][i] + 2.0f * ws[s * reso + i - 1];
          c6 += sv;
          ws[(3 + s) * reso + i] = c6;
        }
      }
    }
    return;
  }

  __shared__ float raw[256];      // within-chunk cumsums: raw[n*16+m], n = col
  __shared__ float fin[3][128];   // finalized A3 arrays (offset included)

  const int lane  = threadIdx.x;        // 0..31
  const int mn    = lane & 15;          // row M for A, col N for B
  const int kbase = (lane >> 4) << 1;   // 0 (lanes 0-15) or 2 (lanes 16-31)
  const int mofs  = (lane >> 4) * 8;    // D rows: r + mofs

  for (int phase = 0; phase < 2; ++phase) {      // 0: A3 scans, 1: A6 scans
    for (int batch = 0; batch < 2; ++batch) {    // batch0: seqs {0,1}, batch1: {2,dummy}
      __syncthreads();
      const int s0 = batch * 2;
      const int s1 = batch * 2 + 1;

      // sequence element generator (branchless OOB-safe)
      auto seqv = [&](int seq, int i) -> float {
        if (seq > 2) return 0.0f;                       // dummy columns
        const float* L = (seq == 0) ? xL : (seq == 1) ? yL : zL;
        int im1 = (i > 0) ? (i - 1) : 0;
        float a = L[im1], b = L[i];
        float v;
        if (phase == 0) v = 2.0f * (a + b);
        else            v = 3.0f * a + b + 2.0f * fin[seq][im1];
        return (i == 0) ? 0.0f : v;
      };

      v8f acc = {};
      for (int kk = 0; kk < 4; ++kk) {
        const int k0 = 4 * kk + kbase;
        v2f A, B;
        // A = lower-triangular ones tile: A[m][j] = (j <= m)
        A.x = (k0     <= mn) ? 1.0f : 0.0f;
        A.y = (k0 + 1 <= mn) ? 1.0f : 0.0f;
        // B columns = chunks: col n<8 -> seq s0 chunk n; n>=8 -> seq s1 chunk n-8
        const int seq = (mn >> 3) ? s1 : s0;
        const int ci  = (mn & 7) << 4;
        B.x = seqv(seq, ci + k0);
        B.y = seqv(seq, ci + k0 + 1);
        acc = __builtin_amdgcn_wmma_f32_16x16x4_f32(
            false, A, false, B, (short)0, acc, false, false);
      }

      // D layout: VGPR r holds (M = r + mofs, N = mn)
      for (int r = 0; r < 8; ++r)
        raw[mn * 16 + mofs + r] = acc[r];
      __syncthreads();

      // carry fix-up across chunk totals + write out
      for (int idx = lane; idx < 256; idx += 32) {
        const int sl  = idx >> 7;        // 0/1 within batch
        const int seq = batch * 2 + sl;
        if (seq > 2) continue;
        const int i = idx & 127;
        const int c = i >> 4;
        float v = raw[idx];
        for (int cc = 0; cc < c; ++cc)
          v += raw[(sl << 7) + cc * 16 + 15];   // previous chunk totals
        if (phase == 0) {
          v += offv[seq];
          fin[seq][i] = v;
          ws[seq * 128 + i] = v;                // xA3 / yA4 / zA5
        } else {
          ws[384 + seq * 128 + i] = v;          // xA6 / yA6 / zA6
        }
      }
    }
  }
}

// ---------------------------------------------------------------------------
// Main kernel: HBM-streaming gather. Tables live in LDS (async-staged).
// LDS layout (floats): [0)x [128)y [256)z [384)xA3 [512)yA4 [640)zA5
//                      [768)xA6 [896)yA6 [1024)zA6  -> 1152 floats
// ---------------------------------------------------------------------------
__global__ __launch_bounds__(256) void qg_main_kernel(
    const float* __restrict__ renderP, const int* __restrict__ renderI,
    const float* __restrict__ sdfP,    const int* __restrict__ sdfI,
    const float* __restrict__ xL, const float* __restrict__ yL,
    const float* __restrict__ zL, const float* __restrict__ offv,
    const float* __restrict__ ws, float* __restrict__ out, int N, int reso) {
  __shared__ float sm[1152];
  const int tid    = threadIdx.x;
  const int stride = gridDim.x * blockDim.x;
  const float off3    = offv[3];
  const float invReso = 1.0f / (float)reso;

  if (reso == 128) {
    // ---- async-stage all 9 tables into LDS (1152 B32 transfers/block) ----
#pragma unroll
    for (int it = 0; it < 5; ++it) {
      int j = it * 256 + tid;
      if (j > 1151) j = 1151;                 // uniform trip count, clamped dup
      const float* g;
      if (j < 384) {
        const int seg = j >> 7;
        const float* base = (seg == 0) ? xL : (seg == 1) ? yL : zL;
        g = base + (j & 127);
      } else {
        g = ws + (j - 384);                   // six prefix arrays, contiguous
      }
      async_g2l_b32((uint32_t)(uintptr_t)&sm[j], g);
    }
    wait_async0();
    __syncthreads();

    for (int i = blockIdx.x * blockDim.x + tid; i < N; i += stride) {
      // look-ahead prefetch of the streamed point data
      __builtin_prefetch(&sdfP[3 * (size_t)(i + stride)], 0, 1);
      __builtin_prefetch(&renderP[3 * (size_t)(i + stride)], 0, 1);

      // ---- SDF side ----
      const int idx = __builtin_nontemporal_load(&sdfI[i]);
      const float px = __builtin_nontemporal_load(&sdfP[3 * (size_t)i + 0]);
      const float py = __builtin_nontemporal_load(&sdfP[3 * (size_t)i + 1]);
      const float pz = __builtin_nontemporal_load(&sdfP[3 * (size_t)i + 2]);
      const int iz = idx & 127, iy = (idx >> 7) & 127, ix = idx >> 14;
      const float a0 = sm[iz] * px;
      const float a1 = sm[128 + iy] * py;
      const float a2 = sm[256 + ix] * pz;
      const float a3 = sm[384 + iz], a4 = sm[512 + iy], a5 = sm[640 + ix];
      const float a6 = off3 + sm[768 + iz] + sm[896 + iy] + sm[1024 + ix];
      const float num = (a0 + a3) * px + (a1 + a4) * py + (a2 + a5) * pz + a6;
      const float dx = 2.f * a0 + a3, dy = 2.f * a1 + a4, dz = 2.f * a2 + a5;
      const float den = sqrtf(dx * dx + dy * dy + dz * dz);
      __builtin_nontemporal_store(num / den * invReso, &out[i]);

      // ---- normal side ----
      const int jdx = __builtin_nontemporal_load(&renderI[i]);
      const float rx = __builtin_nontemporal_load(&renderP[3 * (size_t)i + 0]);
      const float ry = __builtin_nontemporal_load(&renderP[3 * (size_t)i + 1]);
      const float rz = __builtin_nontemporal_load(&renderP[3 * (size_t)i + 2]);
      const int jz = jdx & 127, jy = (jdx >> 7) & 127, jx = jdx >> 14;
      const float nx = 2.f * sm[jz] * rx + sm[384 + jz];
      const float ny = 2.f * sm[128 + jy] * ry + sm[512 + jy];
      const float nz = 2.f * sm[256 + jx] * rz + sm[640 + jx];
      const float nn  = sqrtf(nx * nx + ny * ny + nz * nz);
      const float inv = 1.0f / fmaxf(nn, 1e-12f);
      const size_t ob = (size_t)N + 3 * (size_t)i;
      __builtin_nontemporal_store(nx * inv, &out[ob + 0]);
      __builtin_nontemporal_store(ny * inv, &out[ob + 1]);
      __builtin_nontemporal_store(nz * inv, &out[ob + 2]);
    }
  } else {
    // generic fallback: tables from global, div/mod decode
    const float* A3 = ws;            const float* A4 = ws + reso;
    const float* A5 = ws + 2 * reso; const float* X6 = ws + 3 * reso;
    const float* Y6 = ws + 4 * reso; const float* Z6 = ws + 5 * reso;
    for (int i = blockIdx.x * blockDim.x + tid; i < N; i += stride) {
      int idx = sdfI[i];
      int iz = idx % reso; int t = idx / reso; int iy = t % reso; int ix = t / reso;
      float px = sdfP[3 * (size_t)i], py = sdfP[3 * (size_t)i + 1], pz = sdfP[3 * (size_t)i + 2];
      float a0 = xL[iz] * px, a1 = yL[iy] * py, a2 = zL[ix] * pz;
      float a3 = A3[iz], a4 = A4[iy], a5 = A5[ix];
      float a6 = off3 + X6[iz] + Y6[iy] + Z6[ix];
      float num = (a0 + a3) * px + (a1 + a4) * py + (a2 + a5) * pz + a6;
      float dx = 2.f * a0 + a3, dy = 2.f * a1 + a4, dz = 2.f * a2 + a5;
      out[i] = num / sqrtf(dx * dx + dy * dy + dz * dz) * invReso;

      int jdx = renderI[i];
      int jz = jdx % reso; int u = jdx / reso; int jy = u % reso; int jx = u / reso;
      float rx = renderP[3 * (size_t)i], ry = renderP[3 * (size_t)i + 1], rz = renderP[3 * (size_t)i + 2];
      float nx = 2.f * xL[jz] * rx + A3[jz];
      float ny = 2.f * yL[jy] * ry + A4[jy];
      float nz = 2.f * zL[jx] * rz + A5[jx];
      float inv = 1.0f / fmaxf(sqrtf(nx * nx + ny * ny + nz * nz), 1e-12f);
      size_t ob = (size_t)N + 3 * (size_t)i;
      out[ob] = nx * inv; out[ob + 1] = ny * inv; out[ob + 2] = nz * inv;
    }
  }
}

// ---------------------------------------------------------------------------
extern "C" void kernel_launch(void* const* d_in, const int* in_sizes, int n_in,
                              void* d_out, int out_size, void* d_ws, size_t ws_size,
                              hipStream_t stream) {
  const float* renderP = (const float*)d_in[0];
  const int*   renderI = (const int*)d_in[1];
  const float* sdfP    = (const float*)d_in[2];
  const int*   sdfI    = (const int*)d_in[3];
  const float* xL      = (const float*)d_in[4];
  const float* yL      = (const float*)d_in[5];
  const float* zL      = (const float*)d_in[6];
  const float* offv    = (const float*)d_in[7];
  const int N    = in_sizes[1];   // renderIndexList element count
  const int reso = in_sizes[4];   // xLayer length
  float* ws  = (float*)d_ws;
  float* out = (float*)d_out;

  // one-wave setup: WMMA-based prefix scans -> 6*reso floats in d_ws
  qg_setup_kernel<<<1, 32, 0, stream>>>(xL, yL, zL, offv, ws, reso);

  // streaming main kernel: ~4 points per thread to amortize LDS staging
  int blocks = (N + 1023) / 1024;
  if (blocks < 1) blocks = 1;
  if (blocks > 8192) blocks = 8192;
  qg_main_kernel<<<blocks, 256, 0, stream>>>(renderP, renderI, sdfP, sdfI,
                                             xL, yL, zL, offv, ws, out, N, reso);
}